// Net_8383776162019
// MI455X (gfx1250) — compile-verified
//
#include <hip/hip_runtime.h>
#include <hip/hip_bf16.h>

typedef __attribute__((ext_vector_type(2))) float v2f;
typedef __attribute__((ext_vector_type(8))) float v8f;

#define N_GRAPHS 1024

// ---------------------------------------------------------------------------
// K0: init deg (self-loop = 1.0) and out[g] = b2
// ---------------------------------------------------------------------------
__global__ void init_kernel(float* __restrict__ deg, float* __restrict__ out,
                            const float* __restrict__ b2, int n_nodes) {
    int i = blockIdx.x * blockDim.x + threadIdx.x;
    if (i < n_nodes) deg[i] = 1.0f;
    if (i < N_GRAPHS) out[i] = b2[0];
}

// ---------------------------------------------------------------------------
// K1: degree accumulation over edge targets (col)
// ---------------------------------------------------------------------------
__global__ void degree_kernel(const int* __restrict__ col, float* __restrict__ deg,
                              int n_edges) {
    int e = blockIdx.x * blockDim.x + threadIdx.x;
    if (e < n_edges) atomicAdd(&deg[col[e]], 1.0f);
}

// ---------------------------------------------------------------------------
// K2: dis[i] = rsqrt(deg[i]); acc[i] = x[i] * (1/deg[i])  (self-loop message)
// ---------------------------------------------------------------------------
__global__ void dis_selfloop_kernel(float* __restrict__ dis, const float* __restrict__ x,
                                    float* __restrict__ acc, int n_nodes) {
    int i = blockIdx.x * blockDim.x + threadIdx.x;
    if (i >= n_nodes) return;
    float d   = dis[i];                 // currently holds degree (>= 1 always)
    float ds  = __frsqrt_rn(d);
    dis[i]    = ds;
    float inv = ds * ds;                // 1/deg
    float4 xi = ((const float4*)x)[i];
    float4 a;
    a.x = xi.x * inv; a.y = xi.y * inv; a.z = xi.z * inv; a.w = xi.w * inv;
    ((float4*)acc)[i] = a;
}

// ---------------------------------------------------------------------------
// K3: edge scatter of raw 4-float features: acc[col] += x[row]*dis[row]*dis[col]
// ---------------------------------------------------------------------------
__global__ void edge_scatter_kernel(const int* __restrict__ row, const int* __restrict__ col,
                                    const float* __restrict__ x, const float* __restrict__ dis,
                                    float* __restrict__ acc, int n_edges) {
    int e = blockIdx.x * blockDim.x + threadIdx.x;
    if (e >= n_edges) return;
    int r = row[e];
    int c = col[e];
    float norm = dis[r] * dis[c];
    float4 xr = ((const float4*)x)[r];
    float* dst = acc + (size_t)c * 4;
    atomicAdd(dst + 0, xr.x * norm);
    atomicAdd(dst + 1, xr.y * norm);
    atomicAdd(dst + 2, xr.z * norm);
    atomicAdd(dst + 3, xr.w * norm);
}

// ---------------------------------------------------------------------------
// K4: per wave: 16 nodes. h = acc(16x4) @ W1(4x64) + b1 via 4x WMMA f32 16x16x4.
//     Then ELU, dot with W2, butterfly-reduce over columns, atomicAdd per graph.
//
// A layout (ISA 7.12.2, 32-bit 16x4): lanes 0-15 hold {K0,K1}, lanes 16-31 {K2,K3},
//   M = lane % 16 down... M = lane%16 per lane, K in VGPRs.
// B layout (4x16): lane n holds column N=n; lanes 0-15: VGPR0=K0,VGPR1=K1;
//   lanes 16-31: VGPR0=K2,VGPR1=K3 (mirror of A).
// C/D layout: lane = column N, VGPR r = row M=r (lanes 0-15) / M=r+8 (lanes 16-31).
// ---------------------------------------------------------------------------
__global__ void node_wmma_pool_kernel(const float* __restrict__ acc,
                                      const float* __restrict__ W1,  // [4][64]
                                      const float* __restrict__ b1,  // [64]
                                      const float* __restrict__ W2,  // [64]
                                      const int* __restrict__ batch,
                                      float* __restrict__ out,
                                      int n_waves) {
    int wave = (blockIdx.x * blockDim.x + threadIdx.x) >> 5;
    int lane = threadIdx.x & 31;
    if (wave >= n_waves) return;          // uniform per wave -> EXEC stays all-1s

    int base   = wave * 16;
    int m16    = lane & 15;               // column index for B/C/D, row index for A
    int hiHalf = lane >> 4;               // 0: K={0,1}/rows 0-7; 1: K={2,3}/rows 8-15
    int node   = base + m16;
    int k0     = hiHalf * 2;

    // A fragment: acc[node][k0], acc[node][k0+1]
    const float* an = acc + (size_t)node * 4;
    v2f a;
    a.x = an[k0];
    a.y = an[k0 + 1];

    float sum[8];
#pragma unroll
    for (int r = 0; r < 8; ++r) sum[r] = 0.0f;

#pragma unroll
    for (int cc = 0; cc < 4; ++cc) {
        int col = cc * 16 + m16;
        // B fragment: W1[k0][col], W1[k0+1][col]
        v2f b;
        b.x = W1[k0 * 64 + col];
        b.y = W1[(k0 + 1) * 64 + col];
        // Seed C with bias b1[col] broadcast down all rows
        float bias = b1[col];
        v8f c;
#pragma unroll
        for (int r = 0; r < 8; ++r) c[r] = bias;

        v8f d = __builtin_amdgcn_wmma_f32_16x16x4_f32(
            /*neg_a=*/false, a, /*neg_b=*/false, b,
            /*c_mod=*/(short)0, c, /*reuse_a=*/false, /*reuse_b=*/false);

        float w2 = W2[col];
#pragma unroll
        for (int r = 0; r < 8; ++r) {
            float v = d[r];
            float e = v > 0.0f ? v : (__expf(v) - 1.0f);   // ELU
            sum[r] += e * w2;
        }
    }

    // Reduce each row-sum across the 16 lanes of its half (xor 1,2,4,8 stays in half)
#pragma unroll
    for (int off = 1; off < 16; off <<= 1) {
#pragma unroll
        for (int r = 0; r < 8; ++r) sum[r] += __shfl_xor(sum[r], off, 32);
    }

    // 8 lanes per half each commit one node's scalar to its graph accumulator
    if (m16 < 8) {
        int m = m16 + hiHalf * 8;         // row within the 16-node tile
        atomicAdd(&out[batch[base + m]], sum[m16]);
    }
}

// ---------------------------------------------------------------------------
// Launch
// ---------------------------------------------------------------------------
extern "C" void kernel_launch(void* const* d_in, const int* in_sizes, int n_in,
                              void* d_out, int out_size, void* d_ws, size_t ws_size,
                              hipStream_t stream) {
    const float* x    = (const float*)d_in[0];   // [N,4]
    const float* W1   = (const float*)d_in[1];   // [4,64]
    const float* b1   = (const float*)d_in[2];   // [64]
    const float* W2   = (const float*)d_in[3];   // [64,1]
    const float* b2   = (const float*)d_in[4];   // [1]
    const int*   eidx = (const int*)d_in[5];     // [2,E]
    const int*   batch= (const int*)d_in[6];     // [N]
    float*       out  = (float*)d_out;           // [1024]

    const int N = in_sizes[0] / 4;               // 1,000,000 nodes
    const int E = in_sizes[5] / 2;               // 4,000,000 edges
    const int* row = eidx;                       // sources
    const int* col = eidx + E;                   // targets

    // Workspace: dis/deg [N] floats, acc [N*4] floats  (20 MB)
    float* dis = (float*)d_ws;
    float* acc = dis + N;

    const int BT = 256;

    init_kernel<<<(N + BT - 1) / BT, BT, 0, stream>>>(dis, out, b2, N);
    degree_kernel<<<(E + BT - 1) / BT, BT, 0, stream>>>(col, dis, E);
    dis_selfloop_kernel<<<(N + BT - 1) / BT, BT, 0, stream>>>(dis, x, acc, N);
    edge_scatter_kernel<<<(E + BT - 1) / BT, BT, 0, stream>>>(row, col, x, dis, acc, E);

    const int n_waves = (N + 15) / 16;           // 62,500 waves of 16 nodes
    const int threads = n_waves * 32;
    node_wmma_pool_kernel<<<(threads + BT - 1) / BT, BT, 0, stream>>>(
        acc, W1, b1, W2, batch, out, n_waves);
}